// GNNEncoder_7129645712026
// MI455X (gfx1250) — compile-verified
//
#include <hip/hip_runtime.h>
#include <hip/hip_bf16.h>

typedef __attribute__((ext_vector_type(2))) float v2f;
typedef __attribute__((ext_vector_type(8))) float v8f;

// Problem constants (from reference)
static constexpr int cB = 2;
static constexpr int cN = 384;
static constexpr int cD = 128;    // hidden dim == MLP dim == output dim
static constexpr int cM = cB * cN; // 768 rows in every GEMM

// ---------------------------------------------------------------------------
// Generic fp32 WMMA GEMM: C[M=768, N=128] (+)= A[768,K] @ Bm[K,128]
//   bias     : optional [128] column bias
//   rowscale : optional [768] per-row scale applied to bias (for deg_i * mb2)
//   addC     : 1 -> accumulate into existing C
//   relu     : 1 -> relu at store
// One wave computes one 16x16 tile with V_WMMA_F32_16X16X4_F32 over K/4 steps.
// Lane layouts per CDNA5 ISA 7.12.2:
//   A 16x4 : lanes 0-15 hold M=0..15 (K=0,1), lanes 16-31 hold M=0..15 (K=2,3)
//   B 4x16 : vgpr0 = rows K{0|2}, vgpr1 = rows K{1|3}, lane&15 = column
//   C/D    : vgpr r -> row r (lanes 0-15) / row r+8 (lanes 16-31), lane&15 = col
// ---------------------------------------------------------------------------
__global__ void gemm_wmma_f32(const float* __restrict__ A, int K,
                              const float* __restrict__ Bm,
                              const float* __restrict__ bias,
                              const float* __restrict__ rowscale,
                              float* __restrict__ C,
                              int addC, int relu)
{
    const int wave  = threadIdx.x >> 5;            // 8 waves / block
    const int lane  = threadIdx.x & 31;
    const int tile  = blockIdx.x * 8 + wave;       // 48 blocks * 8 = 384 tiles
    const int mt    = tile >> 3;                   // 48 row tiles
    const int nt    = tile & 7;                    // 8 col tiles
    const int m0    = mt * 16;
    const int n0    = nt * 16;
    const int l15   = lane & 15;
    const int khalf = lane >> 4;                   // 0 or 1
    const int n     = n0 + l15;                    // this lane's output column
    const int mbase = m0 + (khalf ? 8 : 0);        // C/D row base for this lane

    v8f c = {};
    if (addC) {
#pragma unroll
        for (int r = 0; r < 8; ++r)
            c[r] = C[(size_t)(mbase + r) * cD + n];
    }

    const float* Arow = A + (size_t)(m0 + l15) * K;  // A row for this lane
    for (int k0 = 0; k0 < K; k0 += 4) {
        const int ka = k0 + 2 * khalf;
        v2f a, b;
        a.x = Arow[ka];
        a.y = Arow[ka + 1];
        b.x = Bm[(size_t)ka * cD + n];
        b.y = Bm[(size_t)(ka + 1) * cD + n];
        c = __builtin_amdgcn_wmma_f32_16x16x4_f32(
                false, a, false, b, (short)0, c, false, false);
    }

    const float bv = bias ? bias[n] : 0.0f;
#pragma unroll
    for (int r = 0; r < 8; ++r) {
        const int mr = mbase + r;
        const float s = rowscale ? rowscale[mr] : 1.0f;
        float y = c[r] + s * bv;
        if (relu) y = y > 0.0f ? y : 0.0f;
        C[(size_t)mr * cD + n] = y;
    }
}

// ---------------------------------------------------------------------------
// Masked pairwise aggregation (message MLP layer-1 restructured):
//   R[b,i,:]  = sum_j adj[b,i,j] * relu(P[b,i,:] + Q[b,j,:] + adj*we + mb1)
//   deg[b,i]  = sum_j adj[b,i,j]
// One block per (b,i) row; 128 threads = one channel each; adj row in LDS;
// zero-adjacency columns are skipped (uniform branch, ~90% of pairs).
// ---------------------------------------------------------------------------
__global__ void pairwise_agg(const float* __restrict__ P,
                             const float* __restrict__ Q,
                             const float* __restrict__ adj,
                             const float* __restrict__ we,
                             const float* __restrict__ mb1,
                             float* __restrict__ R,
                             float* __restrict__ deg)
{
    const int bi = blockIdx.x;          // 0 .. cB*cN-1
    const int b  = bi / cN;
    const int d  = threadIdx.x;         // 0 .. 127

    __shared__ float s_adj[cN];
    for (int j = d; j < cN; j += blockDim.x)
        s_adj[j] = adj[(size_t)bi * cN + j];
    __syncthreads();

    const float p  = P[(size_t)bi * cD + d];
    const float w  = we[d];
    const float b1 = mb1[d];
    const float* Qb = Q + (size_t)b * cN * cD;

    float acc = 0.0f;
    float dg  = 0.0f;
    for (int j = 0; j < cN; ++j) {
        const float a = s_adj[j];       // uniform across block
        if (a != 0.0f) {
            float t = p + Qb[(size_t)j * cD + d] + a * w + b1;
            t = t > 0.0f ? t : 0.0f;
            acc += a * t;
            dg  += a;
        }
    }
    R[(size_t)bi * cD + d] = acc;
    if (d == 0) deg[bi] = dg;
}

// ---------------------------------------------------------------------------
// LayerNorm with optional residual add and trailing relu.
// One block per row (128 threads), LDS tree reductions for mean/var.
// ---------------------------------------------------------------------------
__global__ void ln_res_relu(const float* __restrict__ xin,
                            const float* __restrict__ hn,
                            const float* __restrict__ g,
                            const float* __restrict__ beta,
                            float* __restrict__ out,
                            int add_residual, int relu_after)
{
    const int row = blockIdx.x;
    const int d   = threadIdx.x;       // 0 .. 127

    float v = hn[(size_t)row * cD + d];
    if (add_residual) v += xin[(size_t)row * cD + d];

    __shared__ float red[cD];
    red[d] = v;
    __syncthreads();
#pragma unroll
    for (int s = cD / 2; s > 0; s >>= 1) {
        if (d < s) red[d] += red[d + s];
        __syncthreads();
    }
    const float mu = red[0] * (1.0f / cD);
    __syncthreads();

    const float cdev = v - mu;
    red[d] = cdev * cdev;
    __syncthreads();
#pragma unroll
    for (int s = cD / 2; s > 0; s >>= 1) {
        if (d < s) red[d] += red[d + s];
        __syncthreads();
    }
    const float var = red[0] * (1.0f / cD);

    float y = cdev * rsqrtf(var + 1e-5f) * g[d] + beta[d];
    if (relu_after) y = y > 0.0f ? y : 0.0f;
    out[(size_t)row * cD + d] = y;
}

// ---------------------------------------------------------------------------
extern "C" void kernel_launch(void* const* d_in, const int* in_sizes, int n_in,
                              void* d_out, int out_size, void* d_ws, size_t ws_size,
                              hipStream_t stream)
{
    (void)in_sizes; (void)n_in; (void)out_size; (void)ws_size;

    const float* h    = (const float*)d_in[0];   // [768, 64]
    const float* adj  = (const float*)d_in[1];   // [2, 384, 384]
    const float* ip_w = (const float*)d_in[2];   // [64, 128]
    const float* ip_b = (const float*)d_in[3];   // [128]
    const float* mw1  = (const float*)d_in[4];   // [3, 257, 128]
    const float* mb1  = (const float*)d_in[5];   // [3, 128]
    const float* mw2  = (const float*)d_in[6];   // [3, 128, 128]
    const float* mb2  = (const float*)d_in[7];   // [3, 128]
    const float* uw1  = (const float*)d_in[8];   // [3, 256, 128]
    const float* ub1  = (const float*)d_in[9];   // [3, 128]
    const float* uw2  = (const float*)d_in[10];  // [3, 128, 128]
    const float* ub2  = (const float*)d_in[11];  // [3, 128]
    const float* ln_g = (const float*)d_in[12];  // [3, 128]
    const float* ln_b = (const float*)d_in[13];  // [3, 128]

    float* ws = (float*)d_ws;
    const size_t S = (size_t)cM * cD;            // 98304 floats
    float* x   = ws + 0 * S;
    float* P   = ws + 1 * S;
    float* Q   = ws + 2 * S;
    float* R   = ws + 3 * S;
    float* agg = ws + 4 * S;
    float* hid = ws + 5 * S;
    float* hn  = ws + 6 * S;
    float* deg = ws + 7 * S;                     // 768 floats

    // x = relu(h @ ip_w + ip_b)
    gemm_wmma_f32<<<48, 256, 0, stream>>>(h, 64, ip_w, ip_b, nullptr, x, 0, 1);

    for (int l = 0; l < 3; ++l) {
        const float* mw1l = mw1 + (size_t)l * 257 * cD;
        const float* mb1l = mb1 + (size_t)l * cD;
        const float* mw2l = mw2 + (size_t)l * cD * cD;
        const float* mb2l = mb2 + (size_t)l * cD;
        const float* uw1l = uw1 + (size_t)l * 256 * cD;
        const float* ub1l = ub1 + (size_t)l * cD;
        const float* uw2l = uw2 + (size_t)l * cD * cD;
        const float* ub2l = ub2 + (size_t)l * cD;

        // P = x @ mw1[:128], Q = x @ mw1[128:256]   (receiver/sender splits)
        gemm_wmma_f32<<<48, 256, 0, stream>>>(x, cD, mw1l,            nullptr, nullptr, P, 0, 0);
        gemm_wmma_f32<<<48, 256, 0, stream>>>(x, cD, mw1l + 128 * cD, nullptr, nullptr, Q, 0, 0);

        // R_i = sum_j adj * relu(P_i + Q_j + adj*w_e + mb1), deg_i = sum_j adj
        pairwise_agg<<<cM, cD, 0, stream>>>(P, Q, adj, mw1l + 256 * cD, mb1l, R, deg);

        // agg = R @ mw2 + deg .outer. mb2
        gemm_wmma_f32<<<48, 256, 0, stream>>>(R, cD, mw2l, mb2l, deg, agg, 0, 0);

        // hid = relu(x @ uw1[:128] + agg @ uw1[128:] + ub1)  (two passes)
        gemm_wmma_f32<<<48, 256, 0, stream>>>(x,   cD, uw1l,            ub1l,    nullptr, hid, 0, 0);
        gemm_wmma_f32<<<48, 256, 0, stream>>>(agg, cD, uw1l + 128 * cD, nullptr, nullptr, hid, 1, 1);

        // hn = hid @ uw2 + ub2
        gemm_wmma_f32<<<48, 256, 0, stream>>>(hid, cD, uw2l, ub2l, nullptr, hn, 0, 0);

        if (l < 2) {
            // x = relu(LN(x + hn))
            ln_res_relu<<<cM, cD, 0, stream>>>(x, hn, ln_g + l * cD, ln_b + l * cD, x, 1, 1);
        } else {
            // out = LN(hn)
            ln_res_relu<<<cM, cD, 0, stream>>>(x, hn, ln_g + l * cD, ln_b + l * cD,
                                               (float*)d_out, 0, 0);
        }
    }
}